// GMI_Pool_89704686944718
// MI455X (gfx1250) — compile-verified
//
#include <hip/hip_runtime.h>

typedef __attribute__((ext_vector_type(2))) float v2f;
typedef __attribute__((ext_vector_type(8))) float v8f;
typedef __attribute__((ext_vector_type(4))) unsigned int v4u;
typedef __attribute__((ext_vector_type(8))) int v8i;
typedef __attribute__((ext_vector_type(4))) int v4i;

#define B_    64
#define NPG_  1024
#define D_    256
#define NEG_  10
#define K_    512
#define N_    (B_ * NPG_)
#define SLOPE_ 0.2f
#define LAMB_  1.0f

// ---------------- Kernel 1: xw = x @ W_bil ---------------------------------
// Block = 256 threads (8 waves). Block covers 128 rows x 64 cols of xw.
// The 256x64 panel of W_bil is staged into LDS once per block by the Tensor
// Data Mover (tensor_load_to_lds, D# built per ISA §8.3-8.4), waited with
// s_wait_tensorcnt. Each wave then computes a 16x64 strip: one A fragment is
// reused across 4 V_WMMA_F32_16X16X4_F32 issues per K step (f32 kept end-to-
// end: top-k selection downstream is sensitive to score perturbation).
__global__ __launch_bounds__(256) void k_gemm_xw(const float* __restrict__ x,
                                                 const float* __restrict__ W,
                                                 float* __restrict__ xw) {
    __shared__ float Wlds[D_ * 64];            // 64 KB: Wlds[k*64 + c]
    const int wave = threadIdx.x >> 5;
    const int lane = threadIdx.x & 31;
    const int rowBlk  = blockIdx.x >> 2;       // N_/128 row blocks
    const int colBase = (blockIdx.x & 3) * 64; // 4 col blocks of 64

#if __has_builtin(__builtin_amdgcn_tensor_load_to_lds)
    if (wave == 0) {
        const unsigned long long ga =
            (unsigned long long)(const void*)(W + colBase);
        const unsigned int ldsAddr =
            (unsigned int)(unsigned long long)(const void*)&Wlds[0];
        // D# group 0: count=1 | lds_addr | global_addr[56:0] | type=2
        v4u g0 = { 0x1u,
                   ldsAddr,
                   (unsigned int)(ga & 0xffffffffull),
                   (unsigned int)((ga >> 32) & 0x01ffffffull) | 0x80000000u };
        // D# group 1: data_size=4B; tensor 256x256 (stride 256); tile 64x256
        v8i g1 = { (int)0x00020000,        // [17:16] data_size = 2 (4 bytes)
                   (int)(256u << 16),      // tensor_dim0[15:0]  @ bits 63:48
                   (int)(256u << 16),      // tensor_dim1[15:0]  @ bits 95:80
                   (int)(64u  << 16),      // tile_dim0 = 64     @ bits 127:112
                   (int)256,               // tile_dim1 = 256    @ bits 143:128
                   (int)256,               // tensor_dim0_stride @ bits 191:160
                   0, 0 };
        v4i z4 = { 0, 0, 0, 0 };
#if __clang_major__ >= 23
        v8i z8 = { 0, 0, 0, 0, 0, 0, 0, 0 };
        __builtin_amdgcn_tensor_load_to_lds(g0, g1, z4, z4, z8, 0);
#else
        __builtin_amdgcn_tensor_load_to_lds(g0, g1, z4, z4, 0);
#endif
        __builtin_amdgcn_s_wait_tensorcnt(0);
    }
#else
    // Fallback: manual staging, thread t copies row k = t (64 floats)
    {
        const int t = threadIdx.x;
        for (int c = 0; c < 64; ++c)
            Wlds[t * 64 + c] = W[(long)t * D_ + colBase + c];
    }
#endif
    __syncthreads();

    const int m  = lane & 15;                  // A row / B col within tile
    const int hi = lane >> 4;                  // half-wave selects K pair
    const long row = (long)rowBlk * 128 + wave * 16 + m;

    v8f acc[4] = {};
    for (int k = 0; k < D_; k += 4) {
        v2f a;  // A (16x4): lanes 0-15 K=k,k+1 ; lanes 16-31 K=k+2,k+3
        a.x = x[row * D_ + k + 2 * hi];
        a.y = x[row * D_ + k + 2 * hi + 1];
        #pragma unroll
        for (int ct = 0; ct < 4; ++ct) {
            v2f b;  // B (4x16) from LDS panel
            const int c = ct * 16 + m;
            b.x = Wlds[(k + 2 * hi) * 64 + c];
            b.y = Wlds[(k + 2 * hi + 1) * 64 + c];
            acc[ct] = __builtin_amdgcn_wmma_f32_16x16x4_f32(
                false, a, false, b, (short)0, acc[ct], false, false);
        }
    }
    #pragma unroll
    for (int ct = 0; ct < 4; ++ct) {
        union { v8f v; float f[8]; } u; u.v = acc[ct];
        #pragma unroll
        for (int v = 0; v < 8; ++v) {
            const long mm = (long)rowBlk * 128 + wave * 16 + v + 8 * hi;
            const int  nn = colBase + ct * 16 + (lane & 15);
            xw[mm * D_ + nn] = u.f[v];
        }
    }
}

// ---------------- helpers --------------------------------------------------
__device__ __forceinline__ float waveRedSum(float v) {
    #pragma unroll
    for (int m = 16; m >= 1; m >>= 1) v += __shfl_xor(v, m, 32);
    return v;
}
__device__ __forceinline__ float softplusf(float x) {
    return fmaxf(x, 0.f) + log1pf(expf(-fabsf(x)));
}

// ---------------- Kernel 2: node scores (wave32 per node) ------------------
__global__ __launch_bounds__(256) void k_score(const float* __restrict__ xw,
                                               const float* __restrict__ h,
                                               const int* __restrict__ neg_idx,
                                               const float* __restrict__ sb1,
                                               const float* __restrict__ sb2,
                                               const float* __restrict__ bb,
                                               float* __restrict__ score) {
    const int wave = threadIdx.x >> 5, lane = threadIdx.x & 31;
    const long n = (long)blockIdx.x * 8 + wave;
    float xr[8];
    #pragma unroll
    for (int i = 0; i < 8; ++i) xr[i] = xw[n * D_ + lane + 32 * i];

    float pos = 0.f;
    #pragma unroll
    for (int i = 0; i < 8; ++i) pos += xr[i] * h[n * D_ + lane + 32 * i];
    pos = waveRedSum(pos);

    const float b = bb[0];
    float negsum = 0.f;
    for (int k = 0; k < NEG_; ++k) {
        const long j = neg_idx[n * NEG_ + k];
        float d = 0.f;
        #pragma unroll
        for (int i = 0; i < 8; ++i) d += xr[i] * h[j * D_ + lane + 32 * i];
        d = waveRedSum(d);
        negsum += d + b + sb2[n * NEG_ + k];
    }
    if (lane == 0) {
        const float p  = pos + b + sb1[n];
        const float nm = negsum / (float)NEG_;
        score[n] = softplusf(p) + softplusf(nm);
    }
}

// ---------------- Kernel 3: per-graph top-K by rank ------------------------
// Rank = #elements strictly greater, or equal with lower index (matches
// jax.lax.top_k descending order + tie-break). rank < K  ==> kept at `rank`.
__global__ __launch_bounds__(1024) void k_topk(const float* __restrict__ score,
                                               int* __restrict__ node_pos,
                                               int* __restrict__ perm) {
    __shared__ float s[NPG_];
    const int b = blockIdx.x, t = threadIdx.x;
    const int node = b * NPG_ + t;
    const float my = score[node];
    s[t] = my;
    __syncthreads();
    int rank = 0;
    for (int j = 0; j < NPG_; ++j) {
        const float v = s[j];
        rank += (v > my) || (v == my && j < t);
    }
    node_pos[node] = (rank < K_) ? rank : -1;
    if (rank < K_) perm[b * K_ + rank] = node;
}

// ---------------- Kernel 4: gather x_kept, s1/s2, batch_kept ---------------
__global__ __launch_bounds__(256) void k_gather(const float* __restrict__ x,
                                                const int* __restrict__ perm,
                                                const float* __restrict__ att,
                                                float* __restrict__ out_x,
                                                float* __restrict__ s1,
                                                float* __restrict__ s2,
                                                float* __restrict__ out_batch) {
    __shared__ float r1[256], r2[256];
    const int r = blockIdx.x;          // 0 .. B*K-1
    const int t = threadIdx.x;         // 0 .. 255 == feature dim
    const long src = perm[r];
    const float v = x[src * D_ + t];
    out_x[(long)r * D_ + t] = v;
    r1[t] = v * att[t];
    r2[t] = v * att[D_ + t];
    __syncthreads();
    for (int s = 128; s > 0; s >>= 1) {
        if (t < s) { r1[t] += r1[t + s]; r2[t] += r2[t + s]; }
        __syncthreads();
    }
    if (t == 0) {
        s1[r] = r1[0];
        s2[r] = r2[0];
        out_batch[r] = (float)(r / K_);   // int output concatenated as float
    }
}

// ---------------- Kernel 5a: W0 = leaky_relu(s1_i + s2_j) ------------------
__global__ __launch_bounds__(256) void k_w0(const float* __restrict__ s1,
                                            const float* __restrict__ s2,
                                            float* __restrict__ Wf) {
    const long idx = (long)blockIdx.x * 256 + threadIdx.x;  // < B*K*K
    const long b = idx >> 18;
    const int  i = (int)((idx >> 9) & (K_ - 1));
    const int  j = (int)(idx & (K_ - 1));
    float w = s1[b * K_ + i] + s2[b * K_ + j];
    w = (w > 0.f) ? w : SLOPE_ * w;
    Wf[idx] = w;
}

// ---------------- Kernel 5b: edge scatter-add ------------------------------
__global__ __launch_bounds__(256) void k_edges(const int* __restrict__ ei,
                                               const float* __restrict__ ea,
                                               const int* __restrict__ node_pos,
                                               float* __restrict__ Wf,
                                               int E) {
    const int e = blockIdx.x * 256 + threadIdx.x;
    if (e >= E) return;
    const int row = ei[e], col = ei[E + e];
    const int g = row >> 10;                 // row / NPG
    if ((col >> 10) != g) return;
    const int pr = node_pos[row], pc = node_pos[col];
    if (pr < 0 || pc < 0) return;
    atomicAdd(&Wf[((long)g * K_ + pr) * K_ + pc], ea[e] * LAMB_);
}

// ---------------- Kernel 5c: row softmax over j (K=512), wave per row ------
__global__ __launch_bounds__(256) void k_softmax(float* __restrict__ Wf) {
    const int wave = threadIdx.x >> 5, lane = threadIdx.x & 31;
    const long row = (long)blockIdx.x * 8 + wave;   // < B*K rows
    float r[16];
    float mx = -__builtin_inff();
    #pragma unroll
    for (int i = 0; i < 16; ++i) {
        r[i] = Wf[row * K_ + lane + 32 * i];
        mx = fmaxf(mx, r[i]);
    }
    #pragma unroll
    for (int m = 16; m >= 1; m >>= 1) mx = fmaxf(mx, __shfl_xor(mx, m, 32));
    float sum = 0.f;
    #pragma unroll
    for (int i = 0; i < 16; ++i) { r[i] = expf(r[i] - mx); sum += r[i]; }
    sum = waveRedSum(sum);
    const float inv = 1.f / sum;
    #pragma unroll
    for (int i = 0; i < 16; ++i) Wf[row * K_ + lane + 32 * i] = r[i] * inv;
}

// ---------------- launch ---------------------------------------------------
extern "C" void kernel_launch(void* const* d_in, const int* in_sizes, int n_in,
                              void* d_out, int out_size, void* d_ws, size_t ws_size,
                              hipStream_t stream) {
    const float* x   = (const float*)d_in[0];
    const int*   ei  = (const int*)  d_in[1];   // (2, E) row-major
    const float* ea  = (const float*)d_in[2];
    const float* h   = (const float*)d_in[4];
    const int*   neg = (const int*)  d_in[5];
    const float* sb1 = (const float*)d_in[6];
    const float* sb2 = (const float*)d_in[7];
    const float* att = (const float*)d_in[8];
    const float* Wb  = (const float*)d_in[9];
    const float* bb  = (const float*)d_in[10];
    const int E = in_sizes[2];

    // outputs concatenated: x_kept (B*K*D), attn (B*K*K), batch_kept (B*K)
    float* out_x    = (float*)d_out;
    float* out_attn = out_x + (long)B_ * K_ * D_;
    float* out_bat  = out_attn + (long)B_ * K_ * K_;

    // workspace: xw (64MB) + score + s1 + s2 + node_pos + perm
    char* ws = (char*)d_ws;
    float* xw    = (float*)ws;  ws += sizeof(float) * (size_t)N_ * D_;
    float* score = (float*)ws;  ws += sizeof(float) * (size_t)N_;
    float* s1    = (float*)ws;  ws += sizeof(float) * (size_t)B_ * K_;
    float* s2    = (float*)ws;  ws += sizeof(float) * (size_t)B_ * K_;
    int* node_pos = (int*)ws;   ws += sizeof(int) * (size_t)N_;
    int* perm     = (int*)ws;

    k_gemm_xw<<<(N_ / 128) * 4, 256, 0, stream>>>(x, Wb, xw);
    k_score<<<N_ / 8, 256, 0, stream>>>(xw, h, neg, sb1, sb2, bb, score);
    k_topk<<<B_, 1024, 0, stream>>>(score, node_pos, perm);
    k_gather<<<B_ * K_, 256, 0, stream>>>(x, perm, att, out_x, s1, s2, out_bat);
    k_w0<<<(B_ * (long)K_ * K_) / 256, 256, 0, stream>>>(s1, s2, out_attn);
    k_edges<<<(E + 255) / 256, 256, 0, stream>>>(ei, ea, node_pos, out_attn, E);
    k_softmax<<<(B_ * K_) / 8, 256, 0, stream>>>(out_attn);
}